// RedGAN_11141145166399
// MI455X (gfx1250) — compile-verified
//
#include <hip/hip_runtime.h>
#include <math.h>

// Problem constants (match reference)
#define N_NODES 50000
#define N_EDGES 800000
#define IN_C    128
#define HID_C   64
#define EMB_C   32
#define N_HEADS 4
#define GAT1_C  (N_HEADS * HID_C)   // 256
#define BN_INV  0.99999500003749938f // 1/sqrt(1+1e-5)

typedef __attribute__((ext_vector_type(2))) float v2f;
typedef __attribute__((ext_vector_type(8))) float v8f;

enum { EP_NONE = 0, EP_BN = 1, EP_BN_RELU = 2, EP_RELU = 3 };

// ---------------------------------------------------------------------------
// WMMA fp32 GEMM: D[16x16 tile] = A1*W1^T (+ A2*W2^T) (+bias) with epilogue.
// A: [rows x K] row-major node features. W: [outC x K] row-major weights
// (so B = W^T, i.e. B[k][n] = W[n][k]).
// One wave (32 lanes) per 16x16 output tile, exact tiling (N = 3125*16).
// 32-bit WMMA fragment layout (ISA 7.12.2):
//   A 16x4 : lane L -> row L&15,  K = 2*(L>>4) + {0,1}   (2 VGPRs)
//   B 4x16 : lane L -> col L&15,  K = 2*(L>>4) + {0,1}   (2 VGPRs)
//   C/D    : acc[v] -> row v + 8*(L>>4), col L&15         (8 VGPRs)
// ---------------------------------------------------------------------------
__global__ __launch_bounds__(32)
void k_wmma_gemm(const float* __restrict__ A1, int K1, const float* __restrict__ W1,
                 const float* __restrict__ A2, int K2, const float* __restrict__ W2,
                 const float* __restrict__ bias,
                 const float* __restrict__ gamma, const float* __restrict__ beta,
                 float* __restrict__ D, int ldD, int mode)
{
    const int lane    = threadIdx.x;        // 0..31, full wave
    const int rowBase = blockIdx.x << 4;
    const int colBase = blockIdx.y << 4;
    const int mi      = lane & 15;          // A-row / B-col index for this lane
    const int kHi     = (lane >> 4) << 1;   // K sub-offset: lanes 16-31 hold K+2,K+3

    v8f acc = {};

    const float* A = A1; const float* W = W1; int K = K1;
    for (int pass = 0; pass < 2; ++pass) {
        if (A != nullptr) {
            const float* arow = A + (size_t)(rowBase + mi) * K + kHi;
            const float* wrow = W + (size_t)(colBase + mi) * K + kHi;
            for (int k = 0; k < K; k += 4) {
                v2f a, b;
                a[0] = arow[k];     a[1] = arow[k + 1];
                b[0] = wrow[k];     b[1] = wrow[k + 1];
                acc = __builtin_amdgcn_wmma_f32_16x16x4_f32(
                        false, a, false, b, (short)0, acc, false, false);
            }
        }
        A = A2; W = W2; K = K2;
    }

    // Epilogue
    const int col    = colBase + mi;
    const int rowOff = (lane >> 4) << 3;
    const float bval = bias  ? bias[col] : 0.0f;
    const float gsc  = gamma ? gamma[col] * BN_INV : 1.0f;
    const float bsh  = beta  ? beta[col] : 0.0f;
#pragma unroll
    for (int v = 0; v < 8; ++v) {
        float y = acc[v] + bval;
        if (mode == EP_BN || mode == EP_BN_RELU) y = y * gsc + bsh;
        if (mode == EP_BN_RELU || mode == EP_RELU) y = fmaxf(y, 0.0f);
        D[(size_t)(rowBase + rowOff + v) * ldD + col] = y;
    }
}

// ---------------------------------------------------------------------------
// Elementwise / sparse helper kernels
// ---------------------------------------------------------------------------
__device__ __forceinline__ void atomicMaxF(float* addr, float val)
{
    // ordered-int trick; works for any mix when initialized to -inf
    if (val >= 0.0f) atomicMax((int*)addr, __float_as_int(val));
    else             atomicMin((unsigned int*)addr, (unsigned int)__float_as_int(val));
}

__global__ void k_fill(float* __restrict__ p, float v, int n)
{
    int i = blockIdx.x * blockDim.x + threadIdx.x;
    if (i < n) p[i] = v;
}

__global__ void k_degree(const int* __restrict__ dst, float* __restrict__ deg, int e_cnt)
{
    int e = blockIdx.x * blockDim.x + threadIdx.x;
    if (e < e_cnt) atomicAdd(&deg[dst[e]], 1.0f);
}

// agg[dst[e]][c] += x[src[e]][c]   (thread = edge*C + c)
__global__ void k_scatter_add(const float* __restrict__ x,
                              const int* __restrict__ src, const int* __restrict__ dst,
                              float* __restrict__ agg, int C, int total)
{
    int idx = blockIdx.x * blockDim.x + threadIdx.x;
    if (idx >= total) return;
    int e = idx / C, c = idx - e * C;
    atomicAdd(&agg[(size_t)dst[e] * C + c], x[(size_t)src[e] * C + c]);
}

__global__ void k_div_deg(float* __restrict__ agg, const float* __restrict__ deg,
                          int C, int total)
{
    int idx = blockIdx.x * blockDim.x + threadIdx.x;
    if (idx >= total) return;
    agg[idx] /= fmaxf(deg[idx / C], 1.0f);
}

// attention logits per (node, head): als = <xp[n,h,:], asrc[h,:]>, ald likewise
__global__ void k_att_scores(const float* __restrict__ xp,
                             const float* __restrict__ asrc, const float* __restrict__ adst,
                             float* __restrict__ als, float* __restrict__ ald,
                             int H, int C, int total)
{
    int i = blockIdx.x * blockDim.x + threadIdx.x;
    if (i >= total) return;
    int node = i / H, h = i - node * H;
    const float* v  = xp + (size_t)node * H * C + (size_t)h * C;
    const float* as = asrc + (size_t)h * C;
    const float* ad = adst + (size_t)h * C;
    float s1 = 0.0f, s2 = 0.0f;
    for (int c = 0; c < C; ++c) { s1 += v[c] * as[c]; s2 += v[c] * ad[c]; }
    als[i] = s1; ald[i] = s2;
}

__device__ __forceinline__ void edge_pair(const int* src, const int* dst,
                                          int e, int e_cnt, int& s, int& d)
{
    if (e < e_cnt) { s = src[e]; d = dst[e]; }
    else           { s = d = e - e_cnt; }     // appended self-loops
}

// pass 1: segment max of leaky_relu(als[s]+ald[d])
__global__ void k_edge_max(const int* __restrict__ src, const int* __restrict__ dst,
                           const float* __restrict__ als, const float* __restrict__ ald,
                           float* __restrict__ m, int H, int e_cnt, int total)
{
    int idx = blockIdx.x * blockDim.x + threadIdx.x;
    if (idx >= total) return;
    int e = idx / H, h = idx - e * H;
    int s, d; edge_pair(src, dst, e, e_cnt, s, d);
    float v  = als[s * H + h] + ald[d * H + h];
    float ev = v > 0.0f ? v : 0.2f * v;
    atomicMaxF(&m[d * H + h], ev);
}

// pass 2: numerators (cached in alpha) and segment sum of exp
__global__ void k_edge_exp(const int* __restrict__ src, const int* __restrict__ dst,
                           const float* __restrict__ als, const float* __restrict__ ald,
                           const float* __restrict__ m, float* __restrict__ den,
                           float* __restrict__ alpha, int H, int e_cnt, int total)
{
    int idx = blockIdx.x * blockDim.x + threadIdx.x;
    if (idx >= total) return;
    int e = idx / H, h = idx - e * H;
    int s, d; edge_pair(src, dst, e, e_cnt, s, d);
    float v  = als[s * H + h] + ald[d * H + h];
    float ev = v > 0.0f ? v : 0.2f * v;
    float ex = expf(ev - m[d * H + h]);
    alpha[idx] = ex;
    atomicAdd(&den[d * H + h], ex);
}

// pass 3: out[d][h*C+c] += xp[s][h*C+c] * alpha/(den+eps)  (thread = edge*H*C + hc)
__global__ void k_edge_aggr(const int* __restrict__ src, const int* __restrict__ dst,
                            const float* __restrict__ xp, const float* __restrict__ alpha,
                            const float* __restrict__ den, float* __restrict__ out,
                            int H, int C, int e_cnt, int total)
{
    int idx = blockIdx.x * blockDim.x + threadIdx.x;
    if (idx >= total) return;
    int HC = H * C;
    int e = idx / HC, hc = idx - e * HC, h = hc / C;
    int s, d; edge_pair(src, dst, e, e_cnt, s, d);
    float a = alpha[e * H + h] / (den[d * H + h] + 1e-16f);
    atomicAdd(&out[(size_t)d * HC + hc], xp[(size_t)s * HC + hc] * a);
}

// y = elu( (x + bias) * gamma*inv + beta )   in place
__global__ void k_bias_bn_elu(float* __restrict__ g, const float* __restrict__ bias,
                              const float* __restrict__ gamma, const float* __restrict__ beta,
                              int C, int total)
{
    int idx = blockIdx.x * blockDim.x + threadIdx.x;
    if (idx >= total) return;
    int c = idx % C;
    float y = (g[idx] + bias[c]) * (gamma[c] * BN_INV) + beta[c];
    g[idx] = y > 0.0f ? y : expm1f(y);
}

__global__ void k_add_bias(float* __restrict__ g, const float* __restrict__ bias,
                           int C, int total)
{
    int idx = blockIdx.x * blockDim.x + threadIdx.x;
    if (idx < total) g[idx] += bias[idx % C];
}

// risk = sigmoid(<t[n], Wp2> + bp2); red = risk > 0.5
__global__ void k_risk(const float* __restrict__ t, const float* __restrict__ Wp2,
                       const float* __restrict__ bp2,
                       float* __restrict__ risk, float* __restrict__ red, int n)
{
    int i = blockIdx.x * blockDim.x + threadIdx.x;
    if (i >= n) return;
    const float* tr = t + (size_t)i * HID_C;
    float s = bp2[0];
    for (int c = 0; c < HID_C; ++c) s += tr[c] * Wp2[c];
    float r = 1.0f / (1.0f + expf(-s));
    risk[i] = r;
    red[i]  = r > 0.5f ? 1.0f : 0.0f;
}

// ---------------------------------------------------------------------------
// Host-side launch sequence
// ---------------------------------------------------------------------------
static inline int cdiv(long long a, int b) { return (int)((a + b - 1) / b); }

extern "C" void kernel_launch(void* const* d_in, const int* in_sizes, int n_in,
                              void* d_out, int out_size, void* d_ws, size_t ws_size,
                              hipStream_t stream)
{
    const int N  = N_NODES;
    const int E  = N_EDGES;
    const int E2 = E + N;                 // edges + self loops (GAT)

    // inputs (setup_inputs dict order)
    const float* x     = (const float*)d_in[0];
    const int*   ei    = (const int*)  d_in[1];
    const int*   src   = ei;
    const int*   dst   = ei + E;
    const float* Wl1   = (const float*)d_in[2];
    const float* bl1   = (const float*)d_in[3];
    const float* Wr1   = (const float*)d_in[4];
    const float* g1w   = (const float*)d_in[5];
    const float* b1w   = (const float*)d_in[6];
    const float* Wl2   = (const float*)d_in[7];
    const float* bl2   = (const float*)d_in[8];
    const float* Wr2   = (const float*)d_in[9];
    const float* g2w   = (const float*)d_in[10];
    const float* b2w   = (const float*)d_in[11];
    const float* Wg1   = (const float*)d_in[12];
    const float* asrc1 = (const float*)d_in[13];
    const float* adst1 = (const float*)d_in[14];
    const float* bg1   = (const float*)d_in[15];
    const float* gg1   = (const float*)d_in[16];
    const float* bb1   = (const float*)d_in[17];
    const float* Wg2   = (const float*)d_in[18];
    const float* asrc2 = (const float*)d_in[19];
    const float* adst2 = (const float*)d_in[20];
    const float* bg2   = (const float*)d_in[21];
    const float* Wp1   = (const float*)d_in[22];
    const float* bp1   = (const float*)d_in[23];
    const float* Wp2   = (const float*)d_in[24];
    const float* bp2   = (const float*)d_in[25];

    // outputs: emb [N,32] | risk [N] | red [N]
    float* out_emb  = (float*)d_out;
    float* out_risk = out_emb + (size_t)N * EMB_C;
    float* out_red  = out_risk + N;

    // workspace carve (with reuse)
    char* wsp = (char*)d_ws; size_t off = 0;
    auto carve = [&](size_t elems) -> float* {
        float* p = (float*)(wsp + off); off += elems * sizeof(float); return p;
    };
    float* deg    = carve(N);
    float* bufA   = carve((size_t)N * 256);   // agg1|h|agg2 -> xp1 -> xp2|g2|t
    float* bufB   = carve((size_t)N * 256);   // GAT1 accumulate / activated
    float* als1   = carve((size_t)N * N_HEADS);
    float* ald1   = carve((size_t)N * N_HEADS);
    float* m1     = carve((size_t)N * N_HEADS);
    float* den1   = carve((size_t)N * N_HEADS);
    float* alpha1 = carve((size_t)E2 * N_HEADS); // reused as alpha2
    float* als2   = carve(N);
    float* ald2   = carve(N);
    float* m2     = carve(N);
    float* den2   = carve(N);
    (void)ws_size; (void)in_sizes; (void)n_in; (void)out_size;

    float* agg1 = bufA;                       // N x 128
    float* h    = bufA + (size_t)N * 128;     // N x 64
    float* agg2 = bufA + (size_t)N * 192;     // N x 64
    float* xp1  = bufA;                       // N x 256 (after emb done)
    float* xp2  = bufA;                       // N x 32  (after GAT1 done)
    float* g2   = bufA + (size_t)N * 32;      // N x 32
    float* t    = bufA + (size_t)N * 64;      // N x 64
    float* alpha2 = alpha1;

    const int TB = 256;
    const dim3 wave(32);

    // ---- SAGE layer 1 ----
    hipMemsetAsync(deg,  0, (size_t)N * sizeof(float), stream);
    hipMemsetAsync(bufA, 0, (size_t)N * 256 * sizeof(float), stream);
    k_degree<<<cdiv(E, TB), TB, 0, stream>>>(dst, deg, E);
    k_scatter_add<<<cdiv((long long)E * IN_C, TB), TB, 0, stream>>>(x, src, dst, agg1, IN_C, E * IN_C);
    k_div_deg<<<cdiv((long long)N * IN_C, TB), TB, 0, stream>>>(agg1, deg, IN_C, N * IN_C);
    k_wmma_gemm<<<dim3(N / 16, HID_C / 16), wave, 0, stream>>>(
        agg1, IN_C, Wl1, x, IN_C, Wr1, bl1, g1w, b1w, h, HID_C, EP_BN_RELU);

    // ---- SAGE layer 2 -> emb (written straight into d_out) ----
    k_scatter_add<<<cdiv((long long)E * HID_C, TB), TB, 0, stream>>>(h, src, dst, agg2, HID_C, E * HID_C);
    k_div_deg<<<cdiv((long long)N * HID_C, TB), TB, 0, stream>>>(agg2, deg, HID_C, N * HID_C);
    k_wmma_gemm<<<dim3(N / 16, EMB_C / 16), wave, 0, stream>>>(
        agg2, HID_C, Wl2, h, HID_C, Wr2, bl2, g2w, b2w, out_emb, EMB_C, EP_BN);

    // ---- GAT layer 1 (heads=4, concat) ----
    k_wmma_gemm<<<dim3(N / 16, GAT1_C / 16), wave, 0, stream>>>(
        out_emb, EMB_C, Wg1, nullptr, 0, nullptr, nullptr, nullptr, nullptr,
        xp1, GAT1_C, EP_NONE);
    k_att_scores<<<cdiv((long long)N * N_HEADS, TB), TB, 0, stream>>>(
        xp1, asrc1, adst1, als1, ald1, N_HEADS, HID_C, N * N_HEADS);
    k_fill<<<cdiv((long long)N * N_HEADS, TB), TB, 0, stream>>>(m1, -INFINITY, N * N_HEADS);
    hipMemsetAsync(den1, 0, (size_t)N * N_HEADS * sizeof(float), stream);
    hipMemsetAsync(bufB, 0, (size_t)N * 256 * sizeof(float), stream);
    k_edge_max <<<cdiv((long long)E2 * N_HEADS, TB), TB, 0, stream>>>(
        src, dst, als1, ald1, m1, N_HEADS, E, E2 * N_HEADS);
    k_edge_exp <<<cdiv((long long)E2 * N_HEADS, TB), TB, 0, stream>>>(
        src, dst, als1, ald1, m1, den1, alpha1, N_HEADS, E, E2 * N_HEADS);
    k_edge_aggr<<<cdiv((long long)E2 * GAT1_C, TB), TB, 0, stream>>>(
        src, dst, xp1, alpha1, den1, bufB, N_HEADS, HID_C, E, E2 * GAT1_C);
    k_bias_bn_elu<<<cdiv((long long)N * GAT1_C, TB), TB, 0, stream>>>(
        bufB, bg1, gg1, bb1, GAT1_C, N * GAT1_C);

    // ---- GAT layer 2 (heads=1, mean==identity) ----
    k_wmma_gemm<<<dim3(N / 16, EMB_C / 16), wave, 0, stream>>>(
        bufB, GAT1_C, Wg2, nullptr, 0, nullptr, nullptr, nullptr, nullptr,
        xp2, EMB_C, EP_NONE);
    k_att_scores<<<cdiv(N, TB), TB, 0, stream>>>(xp2, asrc2, adst2, als2, ald2, 1, EMB_C, N);
    k_fill<<<cdiv(N, TB), TB, 0, stream>>>(m2, -INFINITY, N);
    hipMemsetAsync(den2, 0, (size_t)N * sizeof(float), stream);
    hipMemsetAsync(g2,   0, (size_t)N * EMB_C * sizeof(float), stream);
    k_edge_max <<<cdiv(E2, TB), TB, 0, stream>>>(src, dst, als2, ald2, m2, 1, E, E2);
    k_edge_exp <<<cdiv(E2, TB), TB, 0, stream>>>(src, dst, als2, ald2, m2, den2, alpha2, 1, E, E2);
    k_edge_aggr<<<cdiv((long long)E2 * EMB_C, TB), TB, 0, stream>>>(
        src, dst, xp2, alpha2, den2, g2, 1, EMB_C, E, E2 * EMB_C);
    k_add_bias<<<cdiv((long long)N * EMB_C, TB), TB, 0, stream>>>(g2, bg2, EMB_C, N * EMB_C);

    // ---- risk predictor ----
    k_wmma_gemm<<<dim3(N / 16, HID_C / 16), wave, 0, stream>>>(
        g2, EMB_C, Wp1, nullptr, 0, nullptr, bp1, nullptr, nullptr, t, HID_C, EP_RELU);
    k_risk<<<cdiv(N, TB), TB, 0, stream>>>(t, Wp2, bp2, out_risk, out_red, N);
}